// UnrolledCP_66108136620281
// MI455X (gfx1250) — compile-verified
//
#include <hip/hip_runtime.h>
#include <math.h>

// Problem constants (match reference)
#define BB 2
#define CC 3
#define HH 256
#define WW 256
#define HW (HH * WW)            // 65536
#define CHW (CC * HW)           // 196608
#define NTOT (BB * CHW)         // 393216
#define NU (NTOT * 2)           // 786432
#define BLK 256
#define NBLK (NTOT / BLK)       // 1536
#define BLK_PER_B (CHW / BLK)   // 768
#define K_STEPS 50

typedef __attribute__((ext_vector_type(2))) float v2f;
typedef __attribute__((ext_vector_type(8))) float v8f;

// Full-wave (32-lane) f32 sum through the matrix pipe.
// A holds one value per lane (the other K slot is 0); B is all-ones, which is
// layout-invariant, so D[m,n] = rowsum[m] regardless of A's exact (M,K) map.
// Lanes 0-15 hold rowsums 0..7 in the 8 accumulator VGPRs, lanes 16-31 hold
// rowsums 8..15; one xor-16 shuffle completes the total in every lane.
__device__ __forceinline__ float wave_sum32(float val) {
  v2f a; a.x = val; a.y = 0.0f;
  v2f ones; ones.x = 1.0f; ones.y = 1.0f;
  v8f c = {};
  v8f d = __builtin_amdgcn_wmma_f32_16x16x4_f32(
      /*neg_a=*/false, a, /*neg_b=*/false, ones,
      /*c_mod=*/(short)0, c, /*reuse_a=*/false, /*reuse_b=*/false);
  float s = ((d[0] + d[1]) + (d[2] + d[3])) + ((d[4] + d[5]) + (d[6] + d[7]));
  s += __shfl_xor(s, 16, 32);
  return s;
}

// Init: x = y, xt = 0, u = nabla(y)
__global__ void k_init(const float* __restrict__ y, float* __restrict__ x,
                       float* __restrict__ xt, float* __restrict__ u) {
  const int i = blockIdx.x * BLK + threadIdx.x;
  const int w = i & (WW - 1);
  const int h = (i >> 8) & (HH - 1);
  const int bc = i >> 16;
  const float yc = y[i];
  x[i] = yc;
  xt[i] = 0.0f;
  const float yh = y[i + ((h < HH - 1) ? WW : 0)];
  const float yw = y[i + ((w < WW - 1) ? 1 : 0)];
  const float dh = (h < HH - 1) ? (yh - yc) : 0.0f;
  const float dw = (w < WW - 1) ? (yw - yc) : 0.0f;
  const int ubase = bc * 2 * HW + (i & (HW - 1));
  u[ubase] = dh;
  u[ubase + HW] = dw;
}

// Step stage 1: v = u + sigma * nabla(xt); emit per-block sums of v^2 per dir.
__global__ void k_grad(const float* __restrict__ xt, const float* __restrict__ u,
                       float* __restrict__ v, float* __restrict__ partials,
                       float sigma) {
  const int i = blockIdx.x * BLK + threadIdx.x;
  const int w = i & (WW - 1);
  const int h = (i >> 8) & (HH - 1);
  const int bc = i >> 16;
  const float xc = xt[i];
  const float xh = xt[i + ((h < HH - 1) ? WW : 0)];
  const float xw = xt[i + ((w < WW - 1) ? 1 : 0)];
  const float dh = (h < HH - 1) ? (xh - xc) : 0.0f;
  const float dw = (w < WW - 1) ? (xw - xc) : 0.0f;
  const int ubase = bc * 2 * HW + (i & (HW - 1));
  const float vh = u[ubase] + sigma * dh;
  const float vw = u[ubase + HW] + sigma * dw;
  v[ubase] = vh;
  v[ubase + HW] = vw;

  const float th = wave_sum32(vh * vh);
  const float tw = wave_sum32(vw * vw);

  __shared__ float lh[8], lw[8];
  const int lane = threadIdx.x & 31;
  const int wv = threadIdx.x >> 5;
  if (lane == 0) { lh[wv] = th; lw[wv] = tw; }
  __syncthreads();
  if (threadIdx.x == 0) {
    float ph = 0.0f, pw = 0.0f;
#pragma unroll
    for (int j = 0; j < 8; ++j) { ph += lh[j]; pw += lw[j]; }
    partials[blockIdx.x * 2 + 0] = ph;
    partials[blockIdx.x * 2 + 1] = pw;
  }
}

// Step stage 2: deterministic finalize of the 4 (batch, dir) sums.
__global__ void k_reduce(const float* __restrict__ partials,
                         float* __restrict__ sums) {
  const int b = blockIdx.x >> 1;
  const int dir = blockIdx.x & 1;
  const float* p = partials + b * BLK_PER_B * 2 + dir;
  float s = 0.0f;
#pragma unroll
  for (int j = 0; j < BLK_PER_B / BLK; ++j)  // 3 chunks of 256
    s += p[(threadIdx.x + j * BLK) * 2];
  const float t = wave_sum32(s);
  __shared__ float l[8];
  const int lane = threadIdx.x & 31;
  const int wv = threadIdx.x >> 5;
  if (lane == 0) l[wv] = t;
  __syncthreads();
  if (threadIdx.x == 0) {
    float tot = 0.0f;
#pragma unroll
    for (int j = 0; j < 8; ++j) tot += l[j];
    sums[blockIdx.x] = tot;  // index = b*2 + dir
  }
}

// Step stage 3: u = scale*v; x = (x - tau*div(scale*v) + tau*y)/(1+tau);
//              xt = (1+chi)*x_new - chi*x_old
__global__ void k_update(const float* __restrict__ y, const float* __restrict__ lambd,
                         const float* __restrict__ v, const float* __restrict__ sums,
                         float* __restrict__ x, float* __restrict__ xt,
                         float* __restrict__ u, float tau, float chi) {
  const int i = blockIdx.x * BLK + threadIdx.x;
  const int w = i & (WW - 1);
  const int h = (i >> 8) & (HH - 1);
  const int bc = i >> 16;
  const int b = bc / CC;

  const float el = expf(lambd[0]);
  const float nh = sqrtf(sums[b * 2 + 0]);
  const float nw = sqrtf(sums[b * 2 + 1]);
  const float sch = el / fmaxf(nh, el);
  const float scw = el / fmaxf(nw, el);

  const int ubase = bc * 2 * HW + (i & (HW - 1));
  const float vh = v[ubase];
  const float vw = v[ubase + HW];
  const float uh = sch * vh;
  const float uw = scw * vw;
  u[ubase] = uh;
  u[ubase + HW] = uw;

  // nablaT at (h,w) with the per-(b,dir) scale folded in; clamp neighbor
  // addresses so no lane ever forms an out-of-range pointer.
  const float vhm = v[ubase + ((h > 0) ? -WW : 0)];
  const float vwm = v[ubase + HW + ((w > 0) ? -1 : 0)];
  float div = 0.0f;
  div -= (h < HH - 1) ? uh : 0.0f;
  div += (h > 0) ? sch * vhm : 0.0f;
  div -= (w < WW - 1) ? uw : 0.0f;
  div += (w > 0) ? scw * vwm : 0.0f;

  const float xo = x[i];
  const float z = xo - tau * div;
  const float xn = (z + tau * y[i]) / (1.0f + tau);
  x[i] = xn;
  xt[i] = (1.0f + chi) * xn - chi * xo;
}

extern "C" void kernel_launch(void* const* d_in, const int* in_sizes, int n_in,
                              void* d_out, int out_size, void* d_ws, size_t ws_size,
                              hipStream_t stream) {
  (void)in_sizes; (void)n_in; (void)out_size; (void)ws_size;
  const float* y = (const float*)d_in[0];
  const float* lambd = (const float*)d_in[1];

  // Outputs live in d_out: [x | xt | u]
  float* x = (float*)d_out;
  float* xt = x + NTOT;
  float* u = xt + NTOT;

  // Scratch: raw v, block partials, final 4 sums
  float* v = (float*)d_ws;
  float* partials = v + NU;
  float* sums = partials + NBLK * 2;

  k_init<<<NBLK, BLK, 0, stream>>>(y, x, xt, u);

  const float sigma_init = 0.5f / sqrtf(2.0f * (float)CC);
  float sigma = sigma_init;
  float tau = sigma_init;
  for (int s = 0; s < K_STEPS; ++s) {
    k_grad<<<NBLK, BLK, 0, stream>>>(xt, u, v, partials, sigma);
    k_reduce<<<4, BLK, 0, stream>>>(partials, sums);
    const float chi = 1.0f / sqrtf(1.0f + tau);  // 2*RHO = 1
    k_update<<<NBLK, BLK, 0, stream>>>(y, lambd, v, sums, x, xt, u, tau, chi);
    tau *= chi;
    sigma /= chi;
  }
}